// SparseVertexConvexCollisionGeometry_27839978013348
// MI455X (gfx1250) — compile-verified
//
#include <hip/hip_runtime.h>
#include <hip/hip_bf16.h>

typedef __attribute__((ext_vector_type(2))) float v2f;
typedef __attribute__((ext_vector_type(8))) float v8f;

#define NVERTS 1024
#define NTILES (NVERTS / 16)
#define FLT_BIG 3.402823466e38f   // finite sentinel: med3(a,b,FLT_BIG) == max(a,b), no maxnum fold

// keys are floats whose low 10 mantissa bits carry the vertex index;
// float ordering of the perturbed values is a valid total order (finite, no NaN).
// '+' instead of '|' (equivalent: low bits are zero, vidx < 1024) to invite v_add3 fusion.
__device__ __forceinline__ float pack_keyf(float v, uint32_t vidx) {
    return __uint_as_float((__float_as_uint(v) & 0xFFFFFC00u) + vidx);
}

// insert x into sorted-desc (k0 >= k1 >= k2 >= k3) via the med3 identity:
// new k0 = max(k0,x); new k_i = med3(k_{i-1}, k_i, x)  -- 4 independent ops
__device__ __forceinline__ void topk4_insert(float& k0, float& k1,
                                             float& k2, float& k3, float x) {
    const float o0 = __builtin_amdgcn_fmed3f(k0, x, FLT_BIG);  // max(k0, x)
    const float o1 = __builtin_amdgcn_fmed3f(k0, k1, x);
    const float o2 = __builtin_amdgcn_fmed3f(k1, k2, x);
    const float o3 = __builtin_amdgcn_fmed3f(k2, k3, x);
    k0 = o0; k1 = o1; k2 = o2; k3 = o3;
}

__device__ __forceinline__ float bperm_f(int srcLane, float v) {
    return __int_as_float(__builtin_amdgcn_ds_bpermute(srcLane << 2, __float_as_int(v)));
}

__global__ __launch_bounds__(256)
void svccg_support_topk_solve(const float* __restrict__ dirs,
                              const float* __restrict__ hint,
                              const float* __restrict__ verts,
                              float* __restrict__ out,
                              int nDirs)
{
    // padded AoS-of-4 vertex cache: x,y,z,0 per vertex (16 KB of 320 KB LDS)
    __shared__ float lds4[NVERTS * 4];

    const int tid = threadIdx.x;
    for (int i = tid; i < NVERTS; i += 256) {
        lds4[4 * i + 0] = verts[3 * i + 0];
        lds4[4 * i + 1] = verts[3 * i + 1];
        lds4[4 * i + 2] = verts[3 * i + 2];
        lds4[4 * i + 3] = 0.0f;
    }
    __syncthreads();

    const int lane = tid & 31;
    const int wave = tid >> 5;
    const int lm   = lane & 15;     // matrix column (direction slot)
    const int half = lane >> 4;     // 0: K=0,1 / M=0..7   1: K=2,3 / M=8..15
    const int d    = blockIdx.x * 128 + wave * 16 + lm;   // this lane's direction

    // ---- B operand (4x16, directions): low lanes (x,y), high lanes (z,0) ----
    float b0 = dirs[3 * d + 2 * half];          // x or z
    float b1 = dirs[3 * d + 1];                 // y (loaded by all lanes)
    b1 = half ? 0.0f : b1;                      // K=3 row is zero
    v2f Bm; Bm.x = b0; Bm.y = b1;

    // ---- per-lane running top-4 as index-tagged float keys (desc) ----
    float k0 = -INFINITY, k1 = -INFINITY, k2 = -INFINITY, k3 = -INFINITY;

    // software-pipelined A-operand prefetch (tile 0)
    float2 av = *(const float2*)&lds4[lm * 4 + 2 * half];

    #pragma unroll 4
    for (int t = 0; t < NTILES; ++t) {
        v2f Am; Am.x = av.x; Am.y = av.y;
        v8f acc = {};
        acc = __builtin_amdgcn_wmma_f32_16x16x4_f32(
            /*neg_a=*/false, Am, /*neg_b=*/false, Bm,
            /*c_mod=*/(short)0, acc, /*reuse_a=*/false, /*reuse_b=*/false);

        // prefetch next tile's A while the insert chain runs
        const int tn = (t + 1) & (NTILES - 1);
        av = *(const float2*)&lds4[(tn * 16 + lm) * 4 + 2 * half];

        // lane (col=lm, halfrow=half): acc[r] = dot(vertex t*16 + half*8 + r, dir d)
        const uint32_t ibase = (uint32_t)(t * 16 + half * 8);
        #pragma unroll
        for (int r = 0; r < 8; ++r) {
            topk4_insert(k0, k1, k2, k3, pack_keyf(acc[r], ibase + r));
        }
    }

    // ---- merge partner half-row's top-4 (lane ^ 16): fetch all, then insert ----
    const int p = lane ^ 16;
    const float w0 = bperm_f(p, k0);
    const float w1 = bperm_f(p, k1);
    const float w2 = bperm_f(p, k2);
    const float w3 = bperm_f(p, k3);
    topk4_insert(k0, k1, k2, k3, w0);
    topk4_insert(k0, k1, k2, k3, w1);
    topk4_insert(k0, k1, k2, k3, w2);
    topk4_insert(k0, k1, k2, k3, w3);

    // ---- per-direction least-squares refine (lanes 0..15 of each wave) ----
    if (half == 0) {
        const int idx[4] = { (int)(__float_as_uint(k0) & 0x3FFu),
                             (int)(__float_as_uint(k1) & 0x3FFu),
                             (int)(__float_as_uint(k2) & 0x3FFu),
                             (int)(__float_as_uint(k3) & 0x3FFu) };
        float qx[4], qy[4], qz[4];
        #pragma unroll
        for (int k = 0; k < 4; ++k) {
            const int a4 = idx[k] * 4;
            qx[k] = lds4[a4 + 0];
            qy[k] = lds4[a4 + 1];
            qz[k] = lds4[a4 + 2];
        }
        // gram = A A^T  (A = 3x4 of witness vertices), symmetric 3x3
        float gxx = 0, gxy = 0, gxz = 0, gyy = 0, gyz = 0, gzz = 0;
        #pragma unroll
        for (int k = 0; k < 4; ++k) {
            gxx += qx[k] * qx[k]; gxy += qx[k] * qy[k]; gxz += qx[k] * qz[k];
            gyy += qy[k] * qy[k]; gyz += qy[k] * qz[k]; gzz += qz[k] * qz[k];
        }
        const float hx = hint[3 * d + 0];
        const float hy = hint[3 * d + 1];
        const float hz = hint[3 * d + 2];
        // y = gram^{-1} h via symmetric adjugate
        const float c00 =  gyy * gzz - gyz * gyz;
        const float c01 =  gxz * gyz - gxy * gzz;
        const float c02 =  gxy * gyz - gxz * gyy;
        const float c11 =  gxx * gzz - gxz * gxz;
        const float c12 =  gxy * gxz - gxx * gyz;
        const float c22 =  gxx * gyy - gxy * gxy;
        const float det = gxx * c00 + gxy * c01 + gxz * c02;
        const float rdet = 1.0f / det;
        const float yx = (c00 * hx + c01 * hy + c02 * hz) * rdet;
        const float yy = (c01 * hx + c11 * hy + c12 * hz) * rdet;
        const float yz = (c02 * hx + c12 * hy + c22 * hz) * rdet;
        // sol_q = q . y ; out = sum sol_q * q
        float ox = 0, oy = 0, oz = 0;
        #pragma unroll
        for (int k = 0; k < 4; ++k) {
            const float s = qx[k] * yx + qy[k] * yy + qz[k] * yz;
            ox += s * qx[k]; oy += s * qy[k]; oz += s * qz[k];
        }
        out[3 * d + 0] = ox;
        out[3 * d + 1] = oy;
        out[3 * d + 2] = oz;
    }
}

extern "C" void kernel_launch(void* const* d_in, const int* in_sizes, int n_in,
                              void* d_out, int out_size, void* d_ws, size_t ws_size,
                              hipStream_t stream) {
    const float* dirs  = (const float*)d_in[0];   // (B, 3)
    const float* hint  = (const float*)d_in[1];   // (B, 3)
    const float* verts = (const float*)d_in[2];   // (1024, 3)
    float* out = (float*)d_out;                   // (B, 1, 3)

    const int nDirs = in_sizes[0] / 3;            // 262144
    const int grid  = nDirs / 128;                // 128 directions per 256-thread block

    svccg_support_topk_solve<<<dim3(grid), dim3(256), 0, stream>>>(
        dirs, hint, verts, out, nDirs);
}